// ChamferDistLoss_77129022701900
// MI455X (gfx1250) — compile-verified
//
#include <hip/hip_runtime.h>

typedef __attribute__((ext_vector_type(2))) float v2f;
typedef __attribute__((ext_vector_type(4))) float v4f;
typedef __attribute__((ext_vector_type(8))) float v8f;

#define NPTS  8192
#define NB    4
#define TOTAL (NB * NPTS)            // 32768 points per cloud (all batches)

// Workspace layout (float offsets), 2 MB total:
//   P1: cloud1 packed [x, y, z, |p|^2]        (16 B / point)
//   Q1: cloud1 packed [-2x, -2y, -2z, 1]
//   P2, Q2: same for cloud2
#define P1_OFF 0
#define Q1_OFF (TOTAL * 4)
#define P2_OFF (TOTAL * 8)
#define Q2_OFF (TOTAL * 12)

// ---------------------------------------------------------------------------
// Pack kernel: one thread per (cloud, point). Also zeroes the scalar output
// (runs before the main kernel in stream order; deterministic per call).
// ---------------------------------------------------------------------------
__global__ void chamfer_pack(const float* __restrict__ c1,
                             const float* __restrict__ c2,
                             float* __restrict__ ws,
                             float* __restrict__ out) {
  int tid = blockIdx.x * blockDim.x + threadIdx.x;   // 0 .. 2*TOTAL-1
  if (tid == 0) out[0] = 0.0f;
  int cloud = tid >> 15;            // / TOTAL
  int idx   = tid & (TOTAL - 1);
  const float* src = cloud ? c2 : c1;
  float x = src[idx * 3 + 0];
  float y = src[idx * 3 + 1];
  float z = src[idx * 3 + 2];
  float sq = x * x + y * y + z * z;
  v4f p = {x, y, z, sq};
  v4f q = {-2.0f * x, -2.0f * y, -2.0f * z, 1.0f};
  ((v4f*)(ws + (cloud ? P2_OFF : P1_OFF)))[idx] = p;
  ((v4f*)(ws + (cloud ? Q2_OFF : Q1_OFF)))[idx] = q;
}

// ---------------------------------------------------------------------------
// Main kernel: one wave per (dir, batch, group-of-16 "column" points).
//   dir 0: columns = cloud1 points, rows loop over cloud2  (min over c2)
//   dir 1: columns = cloud2 points, rows loop over cloud1  (min over c1)
// Per 16x16 tile, a single V_WMMA_F32_16X16X4_F32 produces the full squared
// distances: A=[x,y,z,|p|^2], B=[-2x',-2y',-2z',1], C=broadcast(|p'_n|^2).
// Column-min is a per-lane 8-VGPR fmin tree + one lane^16 swap at the end.
// ---------------------------------------------------------------------------
__global__ void __launch_bounds__(256)
chamfer_min(const float* __restrict__ ws, float* __restrict__ out) {
  const int lane  = threadIdx.x & 31;
  const int wid   = (blockIdx.x * blockDim.x + threadIdx.x) >> 5;
  const int group = wid & 511;         // 512 groups of 16 column points
  const int batch = (wid >> 9) & 3;    // 4 batches
  const int dir   = wid >> 11;         // 2 directions

  const float* PA  = ws + (dir ? P1_OFF : P2_OFF);  // row cloud  [x,y,z,sq]
  const float* QB  = ws + (dir ? Q2_OFF : Q1_OFF);  // col cloud  [-2x,-2y,-2z,1]
  const float* PBs = ws + (dir ? P2_OFF : P1_OFF);  // col cloud  (for |p'|^2)

  const int col  = lane & 15;   // column n (B/C/D) or row m (A) within tile
  const int half = lane >> 4;   // K pair select: {0,1} vs {2,3}
  const int base = batch * NPTS;
  const int ncol = base + group * 16 + col;

  // B matrix: fixed 16 column points for this wave.
  // Lane layout (32-bit 4x16 B): lanes 0-15 hold K={0,1} of col n=lane,
  // lanes 16-31 hold K={2,3} of col n=lane-16.
  v2f bmat = ((const v2f*)QB)[ncol * 2 + half];

  // C matrix: |p'_n|^2 broadcast down the column (all 8 row-VGPRs).
  float sqn = PBs[ncol * 4 + 3];
  v8f cacc;
#pragma unroll
  for (int k = 0; k < 8; ++k) cacc[k] = sqn;

  // A matrix tile pointer: lane holds point m = t*16 + (lane&15),
  // K pair selected by lane>>4; v2f element index = point*2 + half.
  const v2f* aptr = ((const v2f*)PA) + (base + col) * 2 + half;

  float vmin = 3.402823466e38f;
#pragma unroll 4
  for (int t = 0; t < 512; ++t) {
    v2f amat = aptr[t * 32];   // advance 16 points (32 v2f) per tile
    // D[m][n] = |p_m - p'_n|^2 for a 16x16 tile, one instruction.
    v8f d = __builtin_amdgcn_wmma_f32_16x16x4_f32(
        false, amat, false, bmat, (short)0, cacc, false, false);
    float m01 = fminf(d[0], d[1]);
    float m23 = fminf(d[2], d[3]);
    float m45 = fminf(d[4], d[5]);
    float m67 = fminf(d[6], d[7]);
    vmin = fminf(vmin, fminf(fminf(m01, m23), fminf(m45, m67)));
  }

  // Combine the two row-halves (lanes L and L+16 hold the same column n).
  vmin = fminf(vmin, __shfl_xor(vmin, 16, 32));
  // Sum the 16 column minima (each 16-lane half now holds identical data).
  vmin += __shfl_xor(vmin, 8, 32);
  vmin += __shfl_xor(vmin, 4, 32);
  vmin += __shfl_xor(vmin, 2, 32);
  vmin += __shfl_xor(vmin, 1, 32);
  if (lane == 0) atomicAdd(out, vmin);
}

// ---------------------------------------------------------------------------
extern "C" void kernel_launch(void* const* d_in, const int* in_sizes, int n_in,
                              void* d_out, int out_size, void* d_ws, size_t ws_size,
                              hipStream_t stream) {
  const float* c1 = (const float*)d_in[0];   // [4, 8192, 3] f32
  const float* c2 = (const float*)d_in[1];   // [4, 8192, 3] f32
  float* out = (float*)d_out;                // scalar f32
  float* ws  = (float*)d_ws;                 // needs 2 MB

  // Pack both clouds (+ zero the output accumulator).
  chamfer_pack<<<(2 * TOTAL) / 256, 256, 0, stream>>>(c1, c2, ws, out);

  // 4096 waves = 4 batches x 2 directions x 512 column-groups.
  // 8 waves per 256-thread block -> 512 blocks.
  chamfer_min<<<512, 256, 0, stream>>>(ws, out);
}